// VRPGNN_44942537786041
// MI455X (gfx1250) — compile-verified
//
#include <hip/hip_runtime.h>
#include <stdint.h>

typedef __attribute__((ext_vector_type(2))) float v2f;
typedef __attribute__((ext_vector_type(8))) float v8f;

// Guaranteed-native FP32 global atomic add (non-returning form -> STOREcnt).
// Inline asm avoids any chance of the backend expanding atomicrmw fadd into a
// CAS loop on the hot scatter path.
__device__ __forceinline__ void atomic_add_f32(float* p, float v) {
#if defined(__gfx1250__)
    asm volatile("global_atomic_add_f32 %0, %1, off"
                 :
                 : "v"((unsigned long long)(uintptr_t)p), "v"(v)
                 : "memory");
#else
    __hip_atomic_fetch_add(p, v, __ATOMIC_RELAXED, __HIP_MEMORY_SCOPE_AGENT);
#endif
}

// ---------------- degree / normalization ----------------

__global__ void gcn_init_deg(float* deg, int n) {
    int i = blockIdx.x * blockDim.x + threadIdx.x;
    if (i < n) deg[i] = 1.0f;  // self-loop contributes 1
}

__global__ void gcn_count_deg(const long long* __restrict__ dst, float* deg, long long E) {
    long long t = (long long)blockIdx.x * blockDim.x + threadIdx.x;
    if (t < E) atomic_add_f32(&deg[dst[t]], 1.0f);
}

__global__ void gcn_dinv(float* deg, int n) {
    int i = blockIdx.x * blockDim.x + threadIdx.x;
    if (i < n) {
        float d = deg[i];
        deg[i] = (d > 0.0f) ? rsqrtf(d) : 0.0f;
    }
}

// ---------------- dense GEMM via fp32 WMMA (16x16x4) ----------------
// C[nrows x 64] = A[nrows x K] @ B[K x 64], row-major, fp32.
// One wave computes one 16x16 tile of C. Block = 256 threads = 8 waves
// = 2 row-tiles x 4 col-tiles (32 rows x 64 cols per block).
template <int K>
__global__ void gcn_gemm_wmma(const float* __restrict__ A,
                              const float* __restrict__ B,
                              float* __restrict__ C, int nrows) {
    const int lane = threadIdx.x & 31;
    const int wave = threadIdx.x >> 5;
    const int rowTile = blockIdx.x * 2 + (wave >> 2);
    const int colTile = wave & 3;
    const int rowBase = rowTile * 16;
    if (rowBase >= nrows) return;  // wave-uniform: EXEC stays all-1s for WMMA

    const int half = lane >> 4;    // 0: K={0,1}  1: K={2,3}
    const int l    = lane & 15;    // A: row index, B: col index

    const float* arow = A + (size_t)(rowBase + l) * K + half * 2;
    const float* bcol = B + (size_t)(half * 2) * 64 + colTile * 16 + l;

    v8f c = {};
#pragma unroll
    for (int kk = 0; kk < K; kk += 4) {
        v2f a, b;
        a.x = arow[kk];
        a.y = arow[kk + 1];
        b.x = bcol[(size_t)kk * 64];
        b.y = bcol[(size_t)kk * 64 + 64];
        c = __builtin_amdgcn_wmma_f32_16x16x4_f32(
                false, a, false, b, (short)0, c, false, false);
    }

    // C/D layout: VGPR g -> lanes 0-15: (M=g, N=l); lanes 16-31: (M=8+g, N=l)
    float* crow = C + (size_t)(rowBase + half * 8) * 64 + colTile * 16 + l;
#pragma unroll
    for (int g = 0; g < 8; ++g) crow[(size_t)g * 64] = c[g];
}

// ---------------- aggregation ----------------

// agg[i][f] = h[i][f] * dinv[i]^2   (self-loop edge, also zero-initializes agg)
__global__ void gcn_selfloop_init(const float* __restrict__ h,
                                  const float* __restrict__ dinv,
                                  float* __restrict__ agg, int n) {
    long long t = (long long)blockIdx.x * blockDim.x + threadIdx.x;
    long long total = (long long)n * 64;
    if (t < total) {
        int node = (int)(t >> 6);
        float s = dinv[node];
        agg[t] = h[t] * s * s;
    }
}

// One thread per (edge, float4 chunk): 16 lanes cover one 256B h[src] row.
// agg (25.6MB) and h (25.6MB) are L2-resident (192MB L2); atomics are native
// global_atomic_add_f32 resolved at L2.
__global__ void gcn_scatter(const long long* __restrict__ src,
                            const long long* __restrict__ dst,
                            const float* __restrict__ dinv,
                            const float* __restrict__ h,
                            float* __restrict__ agg, long long E) {
    long long t = (long long)blockIdx.x * blockDim.x + threadIdx.x;
    long long total = E << 4;
    if (t >= total) return;
    long long e = t >> 4;
    int f4 = (int)(t & 15);
    long long s = src[e];
    long long d = dst[e];
    float nrm = dinv[s] * dinv[d];
    float4 v = *((const float4*)(h + s * 64) + f4);
    float* ap = agg + d * 64 + f4 * 4;
    atomic_add_f32(ap + 0, v.x * nrm);
    atomic_add_f32(ap + 1, v.y * nrm);
    atomic_add_f32(ap + 2, v.z * nrm);
    atomic_add_f32(ap + 3, v.w * nrm);
}

__global__ void gcn_bias_relu(const float* __restrict__ agg,
                              const float* __restrict__ bias,
                              float* __restrict__ out, int n) {
    long long t = (long long)blockIdx.x * blockDim.x + threadIdx.x;
    long long total = (long long)n * 64;
    if (t < total) {
        float v = agg[t] + bias[(int)(t & 63)];
        out[t] = v > 0.0f ? v : 0.0f;
    }
}

// scores[i] = dot(h[i,:64], Wout) + bout
__global__ void gcn_out(const float* __restrict__ h,
                        const float* __restrict__ Wout,
                        const float* __restrict__ bout,
                        float* __restrict__ out, int n) {
    int i = blockIdx.x * blockDim.x + threadIdx.x;
    if (i >= n) return;
    float acc = bout[0];
    const float4* hp = (const float4*)(h + (size_t)i * 64);
    const float4* wp = (const float4*)Wout;
#pragma unroll
    for (int j = 0; j < 16; ++j) {
        float4 a = hp[j], w = wp[j];
        acc += a.x * w.x + a.y * w.y + a.z * w.z + a.w * w.w;
    }
    out[i] = acc;
}

// ---------------- launch ----------------

extern "C" void kernel_launch(void* const* d_in, const int* in_sizes, int n_in,
                              void* d_out, int out_size, void* d_ws, size_t ws_size,
                              hipStream_t stream) {
    const float*     x    = (const float*)d_in[0];
    const long long* ei   = (const long long*)d_in[1];
    const float*     W1   = (const float*)d_in[2];
    const float*     b1   = (const float*)d_in[3];
    const float*     W2   = (const float*)d_in[4];
    const float*     b2   = (const float*)d_in[5];
    const float*     Wout = (const float*)d_in[6];
    const float*     bout = (const float*)d_in[7];
    float*           scores = (float*)d_out;

    const int Fin = 128;
    const int N = in_sizes[0] / Fin;
    const long long E = in_sizes[1] / 2;
    const long long* src = ei;
    const long long* dst = ei + E;

    float* ws   = (float*)d_ws;
    float* dinv = ws;                        // N
    float* bufA = ws + N;                    // N*64
    float* bufB = bufA + (size_t)N * 64;     // N*64

    const int TB = 256;
    const int gN   = (N + TB - 1) / TB;
    const long long gE   = (E + TB - 1) / TB;
    const long long gNF  = ((long long)N * 64 + TB - 1) / TB;
    const long long gEF  = ((E << 4) + TB - 1) / TB;
    const int gGemm = (N + 31) / 32;         // 32 rows per block

    // --- normalization ---
    gcn_init_deg<<<gN, TB, 0, stream>>>(dinv, N);
    gcn_count_deg<<<(int)gE, TB, 0, stream>>>(dst, dinv, E);
    gcn_dinv<<<gN, TB, 0, stream>>>(dinv, N);

    // --- layer 1: h = relu(Agg(x @ W1) + b1) ---
    gcn_gemm_wmma<128><<<gGemm, TB, 0, stream>>>(x, W1, bufA, N);
    gcn_selfloop_init<<<(int)gNF, TB, 0, stream>>>(bufA, dinv, bufB, N);
    gcn_scatter<<<(int)gEF, TB, 0, stream>>>(src, dst, dinv, bufA, bufB, E);
    gcn_bias_relu<<<(int)gNF, TB, 0, stream>>>(bufB, b1, bufA, N);

    // --- layer 2: h = relu(Agg(h @ W2) + b2) ---
    gcn_gemm_wmma<64><<<gGemm, TB, 0, stream>>>(bufA, W2, bufB, N);
    gcn_selfloop_init<<<(int)gNF, TB, 0, stream>>>(bufB, dinv, bufA, N);
    gcn_scatter<<<(int)gEF, TB, 0, stream>>>(src, dst, dinv, bufB, bufA, E);
    gcn_bias_relu<<<(int)gNF, TB, 0, stream>>>(bufA, b2, bufB, N);

    // --- output head ---
    gcn_out<<<gN, TB, 0, stream>>>(bufB, Wout, bout, scores, N);
}